// FlowLayer_22333829939774
// MI455X (gfx1250) — compile-verified
//
#include <hip/hip_runtime.h>
#include <math.h>

// ---------------- problem constants ----------------
#define Cc   17
#define CP   20          // channels padded to multiple of 4 (WMMA K-chunks)
#define Bb   32
#define Hh   96
#define Ww   72
#define HW   (Hh*Ww)               // 6912
#define BHW  (Bb*HW)               // 221184
#define NTOT (Bb*Cc*HW)            // 3760128
#define NITER 10
#define EPSf 1e-12f
#define BNEPS 1e-5f
#define FLT_BIG 3.402823466e+38f

// ---------------- workspace layout (float offsets) ----------------
#define WS_COEF   0            // 4 floats: a_c, b_c, a_n, b_n   (x = a*v + b)
#define WS_BN     8            // 68 floats: per-channel sum, sumsq
#define WS_PMIN_C 128
#define WS_PMAX_C (128+256)
#define WS_PMIN_N (128+512)
#define WS_PMAX_N (128+768)
#define WS_YSUM   2048                          // BHW floats (normalized channel-sum of y)
#define WS_G2X    ((size_t)WS_YSUM + BHW)       // NTOT each, column-major layout
#define WS_G2Y    (WS_G2X + (size_t)NTOT)
#define WS_RC     (WS_G2Y + (size_t)NTOT)
#define WS_U1     (WS_RC  + (size_t)NTOT)
#define WS_U2     (WS_U1  + (size_t)NTOT)
// column layout: idx = ((b*W + w)*C + c)*H + h  -> each (b,w) column is 1632 contiguous floats

typedef __attribute__((ext_vector_type(2))) float v2f;
typedef __attribute__((ext_vector_type(8))) float v8f;

// ================= stage 1: per-block min/max partials =================
__global__ void k_minmax_partial(const float* __restrict__ cur,
                                 const float* __restrict__ nxt,
                                 float* __restrict__ ws) {
    const bool isN = blockIdx.x >= 256;
    const float* src = isN ? nxt : cur;
    float* pmin = ws + (isN ? WS_PMIN_N : WS_PMIN_C);
    float* pmax = ws + (isN ? WS_PMAX_N : WS_PMAX_C);
    const int blk = blockIdx.x & 255;
    const int tid = threadIdx.x;
    float mn = FLT_BIG, mx = -FLT_BIG;
    for (int i = blk*256 + tid; i < NTOT; i += 256*256) {
        float v = src[i];
        mn = fminf(mn, v); mx = fmaxf(mx, v);
    }
    __shared__ float smn[256], smx[256];
    smn[tid] = mn; smx[tid] = mx;
    __syncthreads();
    for (int s = 128; s > 0; s >>= 1) {
        if (tid < s) { smn[tid] = fminf(smn[tid], smn[tid+s]);
                       smx[tid] = fmaxf(smx[tid], smx[tid+s]); }
        __syncthreads();
    }
    if (tid == 0) { pmin[blk] = smn[0]; pmax[blk] = smx[0]; }
}

// ================= stage 2: finalize min/max -> normalization coefs =================
__global__ void k_minmax_final(float* __restrict__ ws) {
    __shared__ float s0[256], s1[256], s2[256], s3[256];
    const int t = threadIdx.x;
    s0[t] = ws[WS_PMIN_C + t]; s1[t] = ws[WS_PMAX_C + t];
    s2[t] = ws[WS_PMIN_N + t]; s3[t] = ws[WS_PMAX_N + t];
    __syncthreads();
    for (int s = 128; s > 0; s >>= 1) {
        if (t < s) {
            s0[t] = fminf(s0[t], s0[t+s]); s1[t] = fmaxf(s1[t], s1[t+s]);
            s2[t] = fminf(s2[t], s2[t+s]); s3[t] = fmaxf(s3[t], s3[t+s]);
        }
        __syncthreads();
    }
    if (t == 0) {
        // x = 255*(v - mn)/(mn - mx)  (faithful: negative scale)
        float ac = 255.0f / (s0[0] - s1[0]);
        ws[WS_COEF+0] = ac; ws[WS_COEF+1] = -ac * s0[0];
        float an = 255.0f / (s2[0] - s3[0]);
        ws[WS_COEF+2] = an; ws[WS_COEF+3] = -an * s2[0];
    }
}

// ================= stage 3: normalized channel-sum of y =================
__global__ void k_ysum(const float* __restrict__ nxt, float* __restrict__ ws) {
    const int p = blockIdx.x*256 + threadIdx.x;
    if (p >= BHW) return;
    const float an = ws[WS_COEF+2], bn = ws[WS_COEF+3];
    const int b = p / HW, r = p - b*HW;
    const float* base = nxt + (size_t)b*Cc*HW + r;
    float s = 0.0f;
    #pragma unroll
    for (int c = 0; c < Cc; ++c) s += base[c*HW];
    ws[WS_YSUM + p] = an*s + (float)Cc * bn;
}

// ================= stage 4: g2x, g2y, rho_c in column-major layout =================
__global__ void k_precompute(const float* __restrict__ cur,
                             const float* __restrict__ nxt,
                             float* __restrict__ ws) {
    const int idx = blockIdx.x*256 + threadIdx.x;
    if (idx >= NTOT) return;
    const float ac = ws[WS_COEF+0], bc = ws[WS_COEF+1];
    const float an = ws[WS_COEF+2], bn = ws[WS_COEF+3];
    const int b = idx / (Cc*HW);
    const int r = idx - b*(Cc*HW);
    const int c = r / HW;
    const int s = r - c*HW;
    const int h = s / Ww;
    const int w = s - h*Ww;
    const float x = ac*cur[idx] + bc;
    const float y = an*nxt[idx] + bn;
    const float* ys = ws + WS_YSUM + (size_t)b*HW;
    float g2x, g2y;
    if (w == 0)         g2x = 0.5f*((ac*cur[idx+1]+bc) - x);
    else if (w == Ww-1) g2x = 0.5f*(x - (ac*cur[idx-1]+bc));
    else                g2x = 0.5f*(ys[h*Ww + w + 1] - ys[h*Ww + w - 1]);
    if (h == 0)         g2y = 0.5f*((ac*cur[idx+Ww]+bc) - x);
    else if (h == Hh-1) g2y = 0.5f*(x - (ac*cur[idx-Ww]+bc));
    else                g2y = 0.5f*(ys[(h+1)*Ww + w] - ys[(h-1)*Ww + w]);
    const size_t ci = ((size_t)(b*Ww + w)*Cc + c)*Hh + h;
    ws[WS_G2X + ci] = g2x;
    ws[WS_G2Y + ci] = g2y;
    ws[WS_RC  + ci] = y - x;
}

// ================= stage 5: LDS-resident TV-L1 column solver =================
// One block = one (batch, column). All 10 iterations run out of LDS.
// Channel sums SU[h] = sum_c u[c][h] are the reference's ones-matrix GEMM and are
// computed with V_WMMA_F32_16X16X4_F32 (A = ones 16x4, channels zero-padded to 20).
__global__ __launch_bounds__(128) void k_tvl1(float* __restrict__ ws,
                                              const float* __restrict__ tp,
                                              const float* __restrict__ lp,
                                              const float* __restrict__ ap) {
    const int tid  = threadIdx.x;
    const int lane = tid & 31;
    const int wv   = tid >> 5;
    const size_t colBase = (size_t)blockIdx.x * (Cc*Hh);

    const float tt   = tp[0];
    const float l_t  = lp[0] * tt;
    const float taut = ap[0] / tt;

    __shared__ float sU1[CP*Hh], sU2[CP*Hh];         // padded channels 17..19 stay 0
    __shared__ float sG2X[Cc*Hh], sG2Y[Cc*Hh], sRC[Cc*Hh];
    __shared__ float sP1[Hh], sP2[Hh], sSU1[Hh], sSU2[Hh], sD1[Hh], sD2[Hh];

    const float* gG2X = ws + WS_G2X;
    const float* gG2Y = ws + WS_G2Y;
    const float* gRC  = ws + WS_RC;
    float* gU1 = ws + WS_U1;
    float* gU2 = ws + WS_U2;

    for (int i = tid; i < CP*Hh; i += 128) { sU1[i] = 0.0f; sU2[i] = 0.0f; }
    for (int i = tid; i < Cc*Hh; i += 128) {
        sG2X[i] = gG2X[colBase + i];
        sG2Y[i] = gG2Y[colBase + i];
        sRC[i]  = gRC [colBase + i];
    }
    if (tid < Hh) { sP1[tid] = 0.0f; sP2[tid] = 0.0f; }
    __syncthreads();

    const int n16   = lane & 15;
    const int khalf = (lane >> 4) << 1;    // K placement within B fragment (sum is mapping-invariant)

    for (int it = 0; it < NITER; ++it) {
        // ---- divergence term from previous p (div = 2*C*Dv(P)), folded with tau ----
        if (tid < Hh) {
            const int h = tid;
            float dp1, dp2;
            if (h == 0)          { dp1 =  sP1[0];            dp2 =  sP2[0]; }
            else if (h == Hh-1)  { dp1 = -sP1[Hh-2];         dp2 = -sP2[Hh-2]; }
            else                 { dp1 = sP1[h]-sP1[h-1];    dp2 = sP2[h]-sP2[h-1]; }
            sD1[h] = tt * (2.0f*Cc) * dp1;
            sD2[h] = tt * (2.0f*Cc) * dp2;
        }
        __syncthreads();

        // ---- per-channel u update (pointwise) ----
        for (int i = tid; i < Cc*Hh; i += 128) {
            const int c = i / Hh;
            const int h = i - c*Hh;
            const float gx = sG2X[i], gy = sG2Y[i];
            const float grad = gx*gx + gy*gy + EPSf;
            const float u1o = sU1[i], u2o = sU2[i];
            const float rho = sRC[i] + gx*u1o + gy*u2o + EPSf;
            const float ltg = l_t * grad;
            float coef;
            if (rho < -ltg)        coef = l_t;
            else if (rho > ltg)    coef = -l_t;
            else if (grad > EPSf)  coef = -rho / grad;
            else                   coef = 0.0f;
            sU1[i] = coef*gx + u1o + sD1[h];
            sU2[i] = coef*gy + u2o + sD2[h];
        }
        __syncthreads();

        // ---- channel sums via WMMA ones-GEMM: SU[h] = sum_c u[c][h] ----
        for (int g = wv; g < 6; g += 4) {           // 6 row-groups of 16, 4 waves
            const int hh = g*16 + n16;
            v8f acc1 = {}; v8f acc2 = {};
            v2f ones = {1.0f, 1.0f};
            #pragma unroll
            for (int kc = 0; kc < 5; ++kc) {        // 20 padded channels / K=4
                const int cb = kc*4 + khalf;
                v2f b1 = { sU1[cb*Hh + hh], sU1[(cb+1)*Hh + hh] };
                v2f b2 = { sU2[cb*Hh + hh], sU2[(cb+1)*Hh + hh] };
                acc1 = __builtin_amdgcn_wmma_f32_16x16x4_f32(
                           false, ones, false, b1, (short)0, acc1, false, false);
                acc2 = __builtin_amdgcn_wmma_f32_16x16x4_f32(
                           false, ones, false, b2, (short)0, acc2, false, false);
            }
            if (lane < 16) { sSU1[hh] = acc1[0]; sSU2[hh] = acc2[0]; }
        }
        __syncthreads();

        // ---- forward diff of sums + p update (u1x==u1y structural identity) ----
        if (tid < Hh) {
            const int h = tid;
            const float G1 = (h < Hh-1) ? (sSU1[h+1] - sSU1[h]) : 0.0f;
            const float G2 = (h < Hh-1) ? (sSU2[h+1] - sSU2[h]) : 0.0f;
            const float d1 = 1.0f + taut * sqrtf(2.0f*G1*G1 + EPSf);
            const float d2 = 1.0f + taut * sqrtf(2.0f*G2*G2 + EPSf);
            sP1[h] = (sP1[h] + taut*G1) / d1;
            sP2[h] = (sP2[h] + taut*G2) / d2;
        }
        __syncthreads();
    }

    for (int i = tid; i < Cc*Hh; i += 128) {
        gU1[colBase + i] = sU1[i];
        gU2[colBase + i] = sU2[i];
    }
}

// ================= stage 6: BN statistics (one block per channel) =================
__global__ void k_bn_reduce(float* __restrict__ ws) {
    const int ch = blockIdx.x;                 // 0..33
    const int c  = ch % Cc;
    const float* gU = ws + (ch < Cc ? WS_U1 : WS_U2);
    const int tid = threadIdx.x;
    float sum = 0.0f, sq = 0.0f;
    for (int p = tid; p < BHW; p += 256) {     // iterate (b,w,h): contiguous u reads
        const int b = p / (Ww*Hh);
        const int r = p - b*(Ww*Hh);
        const int w = r / Hh;
        const int h = r - w*Hh;
        const float v = gU[((size_t)(b*Ww + w)*Cc + c)*Hh + h];
        sum += v; sq += v*v;
    }
    __shared__ float ss[256], s2[256];
    ss[tid] = sum; s2[tid] = sq;
    __syncthreads();
    for (int s = 128; s > 0; s >>= 1) {
        if (tid < s) { ss[tid] += ss[tid+s]; s2[tid] += s2[tid+s]; }
        __syncthreads();
    }
    if (tid == 0) { ws[WS_BN + 2*ch] = ss[0]; ws[WS_BN + 2*ch + 1] = s2[0]; }
}

// ================= stage 7: BN apply, write output in [B,2C,H,W] =================
__global__ void k_bn_apply(const float* __restrict__ ws,
                           const float* __restrict__ gamma,
                           const float* __restrict__ beta,
                           float* __restrict__ out) {
    const int o = blockIdx.x*256 + threadIdx.x;
    if (o >= 2*NTOT) return;
    const int b  = o / (2*Cc*HW);
    const int r  = o - b*(2*Cc*HW);
    const int ch = r / HW;
    const int s  = r - ch*HW;
    const int h  = s / Ww;
    const int w  = s - h*Ww;
    const int c  = ch % Cc;
    const float* gU = ws + (ch < Cc ? WS_U1 : WS_U2);
    const float v = gU[((size_t)(b*Ww + w)*Cc + c)*Hh + h];
    const float inv = 1.0f / (float)BHW;
    const float mean = ws[WS_BN + 2*ch] * inv;
    const float var  = ws[WS_BN + 2*ch + 1] * inv - mean*mean;
    out[o] = gamma[ch]*(v - mean)*rsqrtf(var + BNEPS) + beta[ch];
}

// ================= launcher =================
extern "C" void kernel_launch(void* const* d_in, const int* in_sizes, int n_in,
                              void* d_out, int out_size, void* d_ws, size_t ws_size,
                              hipStream_t stream) {
    const float* cur   = (const float*)d_in[0];
    const float* nxt   = (const float*)d_in[1];
    const float* t     = (const float*)d_in[8];
    const float* l     = (const float*)d_in[9];
    const float* a     = (const float*)d_in[10];
    const float* gamma = (const float*)d_in[11];
    const float* beta  = (const float*)d_in[12];
    float* ws  = (float*)d_ws;
    float* out = (float*)d_out;

    k_minmax_partial<<<512, 256, 0, stream>>>(cur, nxt, ws);
    k_minmax_final  <<<1,   256, 0, stream>>>(ws);
    k_ysum          <<<(BHW  + 255)/256, 256, 0, stream>>>(nxt, ws);
    k_precompute    <<<(NTOT + 255)/256, 256, 0, stream>>>(cur, nxt, ws);
    k_tvl1          <<<Bb*Ww, 128, 0, stream>>>(ws, t, l, a);
    k_bn_reduce     <<<2*Cc, 256, 0, stream>>>(ws);
    k_bn_apply      <<<(2*NTOT + 255)/256, 256, 0, stream>>>(ws, gamma, beta, out);
}